// SpectralConv2d_19301583028792
// MI455X (gfx1250) — compile-verified
//
#include <hip/hip_runtime.h>
#include <math.h>

// ---------------------------------------------------------------------------
// SpectralConv2d (Hartley) for MI455X / gfx1250, wave32 + f32 WMMA.
//
// Math (after removing dead code / zero blocks in the reference):
//   T[b,i,r,v]   = sum_y x[b,i,row(r),y] * cas256[y,v]          (r in 0..63)
//   R[b,o,r,v]   = sum_i T[b,i,r,v] * w_g[i,o,u,v]              (g = r>=32)
//   out[b,o,row(r),y] = (sum_v R[b,o,r,v] * M2[v,y]) * 2^-42
//   all other output rows are exactly zero.
// where row(r) = r (r<32) or r+192 (r>=32), and
//   M2 = cas32 @ cas256[:32,:]   (the two trailing idht's composed),
//   2^-42 = 1/(8*32*32*32) * 1/(8*32*256*256).
//
// B-side cas tables are stored TRANSPOSED so each lane's WMMA B-fragment
// (a K,K+1 pair at fixed column) is a contiguous 8-byte load -> b64.
// ---------------------------------------------------------------------------

typedef __attribute__((ext_vector_type(2))) float v2f;
typedef __attribute__((ext_vector_type(8))) float v8f;

#define PI2 6.283185307179586f

// ws layout (floats)
#define CASAT_OFF 0         // casAT[v][y], 32x256   (= cas256[y][v] transposed)
#define M2T_OFF   8192      // M2T[y][v],   256x32   (= M2[v][y] transposed)
#define T_OFF     16384     // T[(b*32+i)*64 + r][v], 16384x32
#define R_OFF     540672    // R[(b*32+o)*64 + r][v], 16384x32

// ---------------------------------------------------------------------------
// Kernel P: build cas tables with exact integer angle reduction (mod N).
// ---------------------------------------------------------------------------
__global__ void __launch_bounds__(256)
precompute_cas(float* __restrict__ casAT, float* __restrict__ M2T) {
    const int idx = blockIdx.x * blockDim.x + threadIdx.x;
    const float s256 = PI2 / 256.0f;
    const float s32  = PI2 / 32.0f;
    if (idx < 8192) {
        // casAT[v*256+y] = cas(2*pi*y*v/256)
        const int v = idx >> 8, y = idx & 255;
        const float a = s256 * (float)((y * v) & 255);
        casAT[idx] = cosf(a) + sinf(a);
    } else if (idx < 16384) {
        // M2T[y*32+v] = sum_t cas32(v,t) * cas256(t,y)
        const int j = idx - 8192;
        const int y = j >> 5, v = j & 31;
        float acc = 0.0f;
        for (int t = 0; t < 32; ++t) {
            const float a1 = s32  * (float)((v * t) & 31);
            const float a2 = s256 * (float)((t * y) & 255);
            acc = fmaf(cosf(a1) + sinf(a1), cosf(a2) + sinf(a2), acc);
        }
        M2T[j] = acc;
    }
}

// ---------------------------------------------------------------------------
// Kernel Z: zero the full output (75% of it stays zero).
// ---------------------------------------------------------------------------
__global__ void __launch_bounds__(256)
zero_fill(float* __restrict__ out, int n) {
    const int idx = (blockIdx.x * blockDim.x + threadIdx.x) * 4;
    if (idx < n) {
        *(float4*)(out + idx) = make_float4(0.f, 0.f, 0.f, 0.f);
    }
}

// ---------------------------------------------------------------------------
// Kernel 1: T = x_rows @ cas256[:, :32].  M=16384, K=256, N=32.
// One wave per 16x32 output tile; 64 K-steps x 2 N-tiles = 128 wmma/wave.
// A and B fragments are each one contiguous b64 per lane per K-step.
// ---------------------------------------------------------------------------
__global__ void __launch_bounds__(128)
dht_rows_k1(const float* __restrict__ x, const float* __restrict__ casAT,
            float* __restrict__ T) {
    const int lane   = threadIdx.x & 31;
    const int waveId = (blockIdx.x << 2) + (threadIdx.x >> 5);  // 0..1023
    const int m  = lane & 15;
    const int kb = (lane >> 4) << 1;   // K base within 4-wide step (A & B frags)

    // this lane's A-matrix row -> source row of x
    const int row  = waveId * 16 + m;       // (b*32+i)*64 + r
    const int r    = row & 63;
    const int bi   = row >> 6;
    const int xrow = (r < 32) ? r : (r + 192);   // 0..31 or 224..255
    const float* __restrict__ xb = x + (size_t)(bi * 256 + xrow) * 256 + kb;
    // B columns: n = lane&15 for tile0, n+16 for tile1; rows = K (contiguous)
    const float* __restrict__ c0 = casAT + m * 256 + kb;          // N = 0..15
    const float* __restrict__ c1 = casAT + (m + 16) * 256 + kb;   // N = 16..31

    v8f acc0 = {};  // N = 0..15
    v8f acc1 = {};  // N = 16..31
    for (int k0 = 0; k0 < 256; k0 += 4) {
        v2f a;  a.x = xb[k0];  a.y = xb[k0 + 1];
        v2f b0; b0.x = c0[k0]; b0.y = c0[k0 + 1];
        v2f b1; b1.x = c1[k0]; b1.y = c1[k0 + 1];
        acc0 = __builtin_amdgcn_wmma_f32_16x16x4_f32(false, a, false, b0,
                                                     (short)0, acc0, false, false);
        acc1 = __builtin_amdgcn_wmma_f32_16x16x4_f32(false, a, false, b1,
                                                     (short)0, acc1, false, false);
    }
    // C/D layout: VGPR j -> M = j + 8*(lane>>4), N = lane&15
    const int mrowbase = waveId * 16 + ((lane >> 4) << 3);
    const int col = lane & 15;
#pragma unroll
    for (int j = 0; j < 8; ++j) {
        T[(size_t)(mrowbase + j) * 32 + col]      = acc0[j];
        T[(size_t)(mrowbase + j) * 32 + col + 16] = acc1[j];
    }
}

// ---------------------------------------------------------------------------
// Kernel 2: channel mix.  R[b,o,r,v] = sum_i T[b,i,r,v] * w_g[i,o,u,v].
// M=8 per tiny GEMM -> VALU FMAs; T (2MB) + w (8MB) are L2-resident.
// ---------------------------------------------------------------------------
__global__ void __launch_bounds__(256)
channel_mix_k2(const float* __restrict__ T, const float* __restrict__ w1,
               const float* __restrict__ w2, float* __restrict__ R) {
    const int idx = blockIdx.x * blockDim.x + threadIdx.x;  // 524288 threads
    const int v = idx & 31;
    const int u = (idx >> 5) & 31;
    const int o = (idx >> 10) & 31;
    const int b = (idx >> 15) & 7;
    const int g = idx >> 18;
    const float* __restrict__ w = g ? w2 : w1;
    const int uv = u * 32 + v;
    float acc = 0.0f;
#pragma unroll 8
    for (int i = 0; i < 32; ++i) {
        const float tv = T[(size_t)((b * 32 + i) * 64 + g * 32 + u) * 32 + v];
        const float wv = w[(size_t)(i * 32 + o) * 1024 + uv];
        acc = fmaf(tv, wv, acc);
    }
    R[(size_t)((b * 32 + o) * 64 + g * 32 + u) * 32 + v] = acc;
}

// ---------------------------------------------------------------------------
// Kernel 3: out_rows = R @ M2 * 2^-42.  M=16384, K=32, N=256.
// One wave per 16x64 tile (4 accumulators); 8 K-steps x 4 = 32 wmma/wave.
// Rows scatter to y-rows {0..31, 224..255} of each [256x256] image.
// ---------------------------------------------------------------------------
__global__ void __launch_bounds__(128)
idht_final_k3(const float* __restrict__ R, const float* __restrict__ M2T,
              float* __restrict__ out) {
    const int lane   = threadIdx.x & 31;
    const int waveId = (blockIdx.x << 2) + (threadIdx.x >> 5);  // 0..4095
    const int mtile  = waveId >> 2;     // 0..1023
    const int ntile  = waveId & 3;      // 64-column slab
    const int kb = (lane >> 4) << 1;

    const float* __restrict__ Ar = R + (size_t)(mtile * 16 + (lane & 15)) * 32 + kb;
    const int cbase = ntile * 64 + (lane & 15);
    // B fragment source: M2T[col][k] -> contiguous (k, k+1) pairs
    const float* __restrict__ Mb0 = M2T + (size_t)(cbase +  0) * 32 + kb;
    const float* __restrict__ Mb1 = M2T + (size_t)(cbase + 16) * 32 + kb;
    const float* __restrict__ Mb2 = M2T + (size_t)(cbase + 32) * 32 + kb;
    const float* __restrict__ Mb3 = M2T + (size_t)(cbase + 48) * 32 + kb;

    v8f acc0 = {}, acc1 = {}, acc2 = {}, acc3 = {};
#pragma unroll
    for (int k0 = 0; k0 < 32; k0 += 4) {
        v2f a;  a.x = Ar[k0];   a.y = Ar[k0 + 1];
        v2f b0; b0.x = Mb0[k0]; b0.y = Mb0[k0 + 1];
        v2f b1; b1.x = Mb1[k0]; b1.y = Mb1[k0 + 1];
        v2f b2; b2.x = Mb2[k0]; b2.y = Mb2[k0 + 1];
        v2f b3; b3.x = Mb3[k0]; b3.y = Mb3[k0 + 1];
        acc0 = __builtin_amdgcn_wmma_f32_16x16x4_f32(false, a, false, b0,
                                                     (short)0, acc0, false, false);
        acc1 = __builtin_amdgcn_wmma_f32_16x16x4_f32(false, a, false, b1,
                                                     (short)0, acc1, false, false);
        acc2 = __builtin_amdgcn_wmma_f32_16x16x4_f32(false, a, false, b2,
                                                     (short)0, acc2, false, false);
        acc3 = __builtin_amdgcn_wmma_f32_16x16x4_f32(false, a, false, b3,
                                                     (short)0, acc3, false, false);
    }
    const float scale = 2.2737367544323206e-13f;  // 2^-42
    const int mrowbase = mtile * 16 + ((lane >> 4) << 3);
#pragma unroll
    for (int j = 0; j < 8; ++j) {
        const int row  = mrowbase + j;      // (b*32+o)*64 + r
        const int r    = row & 63;
        const int bo   = row >> 6;
        const int xrow = (r < 32) ? r : (r + 192);
        float* __restrict__ orow = out + (size_t)(bo * 256 + xrow) * 256;
        orow[cbase +  0] = acc0[j] * scale;
        orow[cbase + 16] = acc1[j] * scale;
        orow[cbase + 32] = acc2[j] * scale;
        orow[cbase + 48] = acc3[j] * scale;
    }
}

// ---------------------------------------------------------------------------
extern "C" void kernel_launch(void* const* d_in, const int* in_sizes, int n_in,
                              void* d_out, int out_size, void* d_ws, size_t ws_size,
                              hipStream_t stream) {
    const float* x  = (const float*)d_in[0];   // [8,32,256,256]
    const float* w1 = (const float*)d_in[1];   // [32,32,32,32]
    const float* w2 = (const float*)d_in[2];   // [32,32,32,32]
    float* out = (float*)d_out;                // [8,32,256,256]
    float* ws  = (float*)d_ws;                 // needs ~4.3 MB

    float* casAT = ws + CASAT_OFF;
    float* M2T   = ws + M2T_OFF;
    float* T     = ws + T_OFF;
    float* R     = ws + R_OFF;

    precompute_cas<<<64, 256, 0, stream>>>(casAT, M2T);
    zero_fill<<<(out_size / 4 + 255) / 256, 256, 0, stream>>>(out, out_size);
    dht_rows_k1<<<256, 128, 0, stream>>>(x, casAT, T);       // 1024 waves
    channel_mix_k2<<<2048, 256, 0, stream>>>(T, w1, w2, R);  // 524288 threads
    idht_final_k3<<<1024, 128, 0, stream>>>(R, M2T, out);    // 4096 waves
}